// NN_26508538151674
// MI455X (gfx1250) — compile-verified
//
#include <hip/hip_runtime.h>

typedef float v2f __attribute__((ext_vector_type(2)));
typedef float v8f __attribute__((ext_vector_type(8)));

#define DIN 16
#define HD 128
#define MT 4                 // 16-sample M tiles per workgroup
#define SAMP (16 * MT)       // 64 samples per workgroup
#define XPAD 20              // x-tile LDS row stride (floats): 16B-aligned float4 stores,
                             // conflict-free strided A-loads (20*m mod 64 distinct, m<16)
#define YSTRIDE 132          // y1 LDS row stride: conflict-free A-loads & C-layout stores

// D = A*B + C with V_WMMA_F32_16X16X4_F32 (fp32 matrix core, wave32)
#define WMMA_F32_K4(A, B, C) \
  __builtin_amdgcn_wmma_f32_16x16x4_f32(false, (A), false, (B), (short)0, (C), false, false)

__global__ __launch_bounds__(256) void pinn_jvp_kernel(
    const float* __restrict__ x, const float* __restrict__ xdot,
    const float* __restrict__ W1, const float* __restrict__ W2,
    const float* __restrict__ W3,
    float* __restrict__ outB, float* __restrict__ outBdot,
    float* __restrict__ outY2)
{
  __shared__ __attribute__((aligned(16))) float xs  [SAMP * XPAD];
  __shared__ __attribute__((aligned(16))) float xds [SAMP * XPAD];
  __shared__ __attribute__((aligned(16))) float y1s [SAMP * YSTRIDE];
  __shared__ __attribute__((aligned(16))) float y1ds[SAMP * YSTRIDE];
  __shared__ float partB [8][SAMP];
  __shared__ float partBd[8][SAMP];

  const int tid  = threadIdx.x;
  const int wave = tid >> 5;            // 0..7 : owns H columns [16*wave, 16*wave+16)
  const int lane = tid & 31;
  const int half = lane >> 4;           // WMMA half-wave
  const int l16  = lane & 15;
  const int n0   = blockIdx.x * SAMP;   // sample tile base
  const int hcol = 16 * wave + l16;     // this lane's H column (B/C/D "N" index)

  // ---- cooperative stage of x / xdot tiles into LDS (64 rows x 16 floats each)
  {
    const int r  = tid >> 2;            // 0..63
    const int c4 = (tid & 3) * 4;       // 0,4,8,12
    const float4 vx  = *(const float4*)(x    + (size_t)(n0 + r) * DIN + c4);
    const float4 vxd = *(const float4*)(xdot + (size_t)(n0 + r) * DIN + c4);
    *(float4*)(&xs [r * XPAD + c4]) = vx;
    *(float4*)(&xds[r * XPAD + c4]) = vxd;
  }

  // ---- B-layout W1 tile: B[k][n] = W1[16w+n][k];  lane holds (K=4c+2*half+{0,1}, N=l16)
  v2f bW1[4];
  {
    const float* w1row = W1 + (size_t)hcol * DIN;
    #pragma unroll
    for (int c = 0; c < 4; ++c)
      bW1[c] = *(const v2f*)(w1row + 4 * c + 2 * half);
  }
  // ---- B-layout W2 slab (this wave's 16 output rows, all K=128): register-resident,
  //      loaded once per workgroup per wave (waves own disjoint 16-row slabs)
  v2f bW2[32];
  {
    const float* w2row = W2 + (size_t)hcol * HD;
    #pragma unroll
    for (int c = 0; c < 32; ++c)
      bW2[c] = *(const v2f*)(w2row + 4 * c + 2 * half);
  }
  const float w3v = W3[hcol];

  __syncthreads();   // xs/xds visible

  // ================= Layer 1: z1 = x W1^T ; z1dot = xdot W1^T (4 M tiles) =====
  for (int t = 0; t < MT; ++t) {
    v8f z1 = {}, z1d = {};
    const int arow = (t * 16 + l16) * XPAD + 2 * half;
    #pragma unroll
    for (int c = 0; c < 4; ++c) {
      v2f aX  = *(const v2f*)(&xs [arow + 4 * c]);   // A[M=l16][K=4c+2h,+1]
      v2f aXd = *(const v2f*)(&xds[arow + 4 * c]);
      z1  = WMMA_F32_K4(aX,  bW1[c], z1);
      z1d = WMMA_F32_K4(aXd, bW1[c], z1d);
    }
    // square activation + tangent; C/D: vgpr v -> sample M = t*16 + v + 8*half
    #pragma unroll
    for (int v = 0; v < 8; ++v) {
      float a = z1[v], ad = z1d[v];
      int m = t * 16 + v + 8 * half;
      y1s [m * YSTRIDE + hcol] = a * a;          // y1
      y1ds[m * YSTRIDE + hcol] = 2.0f * a * ad;  // y1dot
    }
  }
  __syncthreads();   // y1/y1dot visible across waves

  // ================= Layer 2 + output layer, per M tile =======================
  for (int t = 0; t < MT; ++t) {
    v8f z2 = {}, z2d = {};
    const int arow = (t * 16 + l16) * YSTRIDE + 2 * half;
    #pragma unroll
    for (int c = 0; c < 32; ++c) {
      v2f aY  = *(const v2f*)(&y1s [arow + 4 * c]);  // A[M][K=4c+2h,+1]
      v2f aYd = *(const v2f*)(&y1ds[arow + 4 * c]);
      z2  = WMMA_F32_K4(aY,  bW2[c], z2);
      z2d = WMMA_F32_K4(aYd, bW2[c], z2d);
    }

    // activation, streamed y2 store, W3 partials
    float pB[8], pBd[8];
    #pragma unroll
    for (int v = 0; v < 8; ++v) {
      float a = z2[v], ad = z2d[v];
      float y2  = a * a;
      float y2d = 2.0f * a * ad;
      int m = t * 16 + v + 8 * half;
      __builtin_nontemporal_store(y2, &outY2[(size_t)(n0 + m) * HD + hcol]);
      pB [v] = w3v * y2;
      pBd[v] = w3v * y2d;
    }
    // reduce over the 16 H-columns held by each half-wave (lane permutes)
    #pragma unroll
    for (int mask = 1; mask < 16; mask <<= 1) {
      #pragma unroll
      for (int v = 0; v < 8; ++v) {
        pB [v] += __shfl_xor(pB [v], mask, 32);
        pBd[v] += __shfl_xor(pBd[v], mask, 32);
      }
    }
    if (l16 == 0) {   // lane 0: samples t*16+0..7, lane 16: t*16+8..15
      #pragma unroll
      for (int v = 0; v < 8; ++v) {
        int m = t * 16 + v + 8 * half;
        partB [wave][m] = pB[v];
        partBd[wave][m] = pBd[v];
      }
    }
  }
  __syncthreads();

  // combine the 8 wave partials; 64 lanes finalize B / Bdot
  if (tid < SAMP) {
    float sB = 0.0f, sBd = 0.0f;
    #pragma unroll
    for (int w = 0; w < 8; ++w) { sB += partB[w][tid]; sBd += partBd[w][tid]; }
    __builtin_nontemporal_store(sB,  &outB   [n0 + tid]);
    __builtin_nontemporal_store(sBd, &outBdot[n0 + tid]);
  }
}

extern "C" void kernel_launch(void* const* d_in, const int* in_sizes, int n_in,
                              void* d_out, int out_size, void* d_ws, size_t ws_size,
                              hipStream_t stream) {
  const float* x    = (const float*)d_in[0];
  const float* xdot = (const float*)d_in[1];
  const float* W1   = (const float*)d_in[2];
  const float* W2   = (const float*)d_in[3];
  const float* W3   = (const float*)d_in[4];

  const int n = in_sizes[0] / DIN;   // 65536 samples (multiple of 64)

  float* outB    = (float*)d_out;          // [n]
  float* outBdot = outB + n;               // [n]
  float* outY2   = outB + 2 * (size_t)n;   // [n, 128]

  dim3 grid(n / SAMP), block(256);
  hipLaunchKernelGGL(pinn_jvp_kernel, grid, block, 0, stream,
                     x, xdot, W1, W2, W3, outB, outBdot, outY2);
}